// SelectiveSearch_10110353015277
// MI455X (gfx1250) — compile-verified
//
#include <hip/hip_runtime.h>
#include <stdint.h>

namespace {
constexpr int kB = 8, kC = 3, kG = 8, kH = 256, kW = 256, kS = 256;
constexpr int kCB = 32, kTB = 8;
constexpr int kHW = kH * kW;
constexpr int kCHW = kC * kCB;        // 96
constexpr int kTHW = kC * kG * kTB;   // 192
constexpr float kImgSize = 65536.0f;

// workspace layout in 32-bit units
constexpr int RS_OFF = 0;                          // u32 counts, B*S
constexpr int CH_OFF = RS_OFF + kB * kS;           // u32 counts -> normalized f32 (in place)
constexpr int TH_OFF = CH_OFF + kB * kS * kCHW;    // u32 counts -> normalized f32 (in place)
constexpr int BBOX_OFF = TH_OFF + kB * kS * kTHW;  // int32 x4: xmin,ymin,xmax,ymax
constexpr int WS_TOTAL = BBOX_OFF + kB * kS * 4;   // ~2.4 MB

typedef float v2f __attribute__((ext_vector_type(2)));
typedef float v8f __attribute__((ext_vector_type(8)));
} // namespace

// ---------------------------------------------------------------- init
__global__ __launch_bounds__(256) void init_ws_kernel(unsigned* ws) {
  int idx = blockIdx.x * blockDim.x + threadIdx.x;
  if (idx >= WS_TOTAL) return;
  if (idx < BBOX_OFF) {
    ws[idx] = 0u;
  } else {
    int q = (idx - BBOX_OFF) & 3;
    ((int*)ws)[idx] = (q == 0) ? kW : (q == 1) ? kH : 0;
  }
}

// ------------------------------------------- sizes + bbox + color hist
__global__ __launch_bounds__(256) void pixel_kernel(const float* __restrict__ imgs,
                                                    const int* __restrict__ lab,
                                                    unsigned* __restrict__ ws) {
  int p = blockIdx.x * blockDim.x + threadIdx.x;
  if (p >= kB * kHW) return;
  int b = p / kHW, hw = p % kHW;
  int s = lab[p];
  atomicAdd(&ws[RS_OFF + b * kS + s], 1u);
  int x = hw % kW, y = hw / kW;
  int* bb = (int*)ws + BBOX_OFF + (size_t)(b * kS + s) * 4;
  atomicMin(&bb[0], x);
  atomicMin(&bb[1], y);
  atomicMax(&bb[2], x);
  atomicMax(&bb[3], y);
#pragma unroll
  for (int c = 0; c < kC; ++c) {
    float v = imgs[(size_t)(b * kC + c) * kHW + hw];
    int cb = (int)(v * (31.0f / 255.0f));
    atomicAdd(&ws[CH_OFF + (size_t)(b * kS + s) * kCHW + c * kCB + cb], 1u);
  }
}

// ---------------------------- texture hist, LDS-privatized per (b,c,g)
__global__ __launch_bounds__(256) void texture_kernel(const float* __restrict__ grads,
                                                      const int* __restrict__ lab,
                                                      unsigned* __restrict__ ws) {
  __shared__ unsigned hist[kS * kTB]; // 8 KB
  int bcg = blockIdx.x;               // b*(C*G) + c*G + g
  int tile = blockIdx.y;              // 0..7
  int b = bcg / (kC * kG);
  int cg = bcg % (kC * kG);
  for (int i = threadIdx.x; i < kS * kTB; i += 256) hist[i] = 0u;
  __syncthreads();
  const int* labb = lab + (size_t)b * kHW;
  const float* gr = grads + (size_t)bcg * kHW;
  int base = tile * (kHW / 8);
  for (int t = threadIdx.x; t < kHW / 8; t += 256) {
    int p = base + t;
    int s = labb[p];
    int tb = (int)(gr[p] * (float)(kTB - 1));
    atomicAdd(&hist[s * kTB + tb], 1u);
  }
  __syncthreads();
  for (int i = threadIdx.x; i < kS * kTB; i += 256) {
    unsigned v = hist[i];
    if (v) {
      int s = i / kTB, tb = i % kTB;
      atomicAdd(&ws[TH_OFF + (size_t)(b * kS + s) * kTHW + cg * kTB + tb], v);
    }
  }
}

// ------------- normalize hist rows in place (u32 counts -> f32 fracs)
__global__ __launch_bounds__(256) void normalize_kernel(unsigned* __restrict__ ws) {
  int gw = (blockIdx.x * blockDim.x + threadIdx.x) >> 5; // one wave32 per (b,s) row
  int lane = threadIdx.x & 31;
  if (gw >= kB * kS) return;
  unsigned* chu = ws + CH_OFF + (size_t)gw * kCHW;
  float cv[3], csum = 0.f;
#pragma unroll
  for (int t = 0; t < 3; ++t) { cv[t] = (float)chu[lane + 32 * t]; csum += cv[t]; }
#pragma unroll
  for (int m = 16; m >= 1; m >>= 1) csum += __shfl_xor(csum, m, 32);
  float cinv = 1.0f / csum;
  float* chf = (float*)chu;
#pragma unroll
  for (int t = 0; t < 3; ++t) chf[lane + 32 * t] = cv[t] * cinv;

  unsigned* thu = ws + TH_OFF + (size_t)gw * kTHW;
  float tv[6], tsum = 0.f;
#pragma unroll
  for (int t = 0; t < 6; ++t) { tv[t] = (float)thu[lane + 32 * t]; tsum += tv[t]; }
#pragma unroll
  for (int m = 16; m >= 1; m >>= 1) tsum += __shfl_xor(tsum, m, 32);
  float tinv = 1.0f / tsum;
  float* thf = (float*)thu;
#pragma unroll
  for (int t = 0; t < 6; ++t) thf[lane + 32 * t] = tv[t] * tinv;
}

// -------------------------------------------------- pairwise affinity
// One block per 16x16 (i,j) tile. Texture/color tiles staged via
// global_load_async_to_lds; size/fill outer-sum via V_WMMA_F32_16X16X4_F32.
__global__ __launch_bounds__(256) void aff_kernel(const unsigned* __restrict__ ws_u,
                                                  float* __restrict__ out) {
  const int jt = blockIdx.x, it = blockIdx.y, b = blockIdx.z;
  const int i0 = it * 16, j0 = jt * 16;
  const int tid = threadIdx.x;

  __shared__ float thI[16 * 204]; // padded stride: 204 % 64 != 0 -> no bank conflicts
  __shared__ float thJ[16 * 204];
  __shared__ float chI[16 * 100];
  __shared__ float chJ[16 * 100];
  __shared__ float rsI[16], rsJ[16];
  __shared__ float boxI[16][4], boxJ[16][4];

  const float* thF = (const float*)(ws_u + TH_OFF);
  const float* chF = (const float*)(ws_u + CH_OFF);

  // Async-stage 2304 16-byte chunks (9 full-EXEC passes of 256 lanes).
  for (int q = tid; q < 2304; q += 256) {
    const float* src;
    uint32_t dst;
    if (q < 1536) { // texture tiles: 2 x 768 chunks
      int qq = q & 767;
      int row = qq / 48, c4 = (qq % 48) * 4;
      int grow = (q < 768) ? (i0 + row) : (j0 + row);
      src = thF + ((size_t)(b * kS + grow) * kTHW + c4);
      float* base = (q < 768) ? thI : thJ;
      dst = (uint32_t)(uintptr_t)(base + row * 204 + c4);
    } else {        // color tiles: 2 x 384 chunks
      int qq = (q - 1536) & 383;
      int row = qq / 24, c4 = (qq % 24) * 4;
      int grow = (q < 1920) ? (i0 + row) : (j0 + row);
      src = chF + ((size_t)(b * kS + grow) * kCHW + c4);
      float* base = (q < 1920) ? chI : chJ;
      dst = (uint32_t)(uintptr_t)(base + row * 100 + c4);
    }
    asm volatile("global_load_async_to_lds_b128 %0, %1, off"
                 :: "v"(dst), "v"((uint64_t)(uintptr_t)src) : "memory");
  }

  if (tid < 16) {
    int r = tid;
    rsI[r] = (float)ws_u[RS_OFF + b * kS + i0 + r];
    const int* bb = (const int*)(ws_u + BBOX_OFF) + (size_t)(b * kS + i0 + r) * 4;
    float xm = (float)bb[0], ym = (float)bb[1];
    boxI[r][0] = xm; boxI[r][1] = ym;
    boxI[r][2] = (float)bb[2] - xm; boxI[r][3] = (float)bb[3] - ym;
  } else if (tid < 32) {
    int r = tid - 16;
    rsJ[r] = (float)ws_u[RS_OFF + b * kS + j0 + r];
    const int* bb = (const int*)(ws_u + BBOX_OFF) + (size_t)(b * kS + j0 + r) * 4;
    float xm = (float)bb[0], ym = (float)bb[1];
    boxJ[r][0] = xm; boxJ[r][1] = ym;
    boxJ[r][2] = (float)bb[2] - xm; boxJ[r][3] = (float)bb[3] - ym;
  }

  asm volatile("s_wait_asynccnt 0" ::: "memory");
  __syncthreads();

  // Histogram intersection (not bilinear -> VALU fmin+fadd from LDS).
  {
    int i = tid >> 4, j = tid & 15;
    const float4* pi = (const float4*)(thI + i * 204);
    const float4* pj = (const float4*)(thJ + j * 204);
    float tacc = 0.f;
#pragma unroll 4
    for (int k = 0; k < 48; ++k) {
      float4 a = pi[k], c = pj[k];
      tacc += fminf(a.x, c.x) + fminf(a.y, c.y) + fminf(a.z, c.z) + fminf(a.w, c.w);
    }
    const float4* qi = (const float4*)(chI + i * 100);
    const float4* qj = (const float4*)(chJ + j * 100);
    float cacc = 0.f;
#pragma unroll 4
    for (int k = 0; k < 24; ++k) {
      float4 a = qi[k], c = qj[k];
      cacc += fminf(a.x, c.x) + fminf(a.y, c.y) + fminf(a.z, c.z) + fminf(a.w, c.w);
    }
    size_t o = (((size_t)b * kS + (i0 + i)) * kS + (j0 + j)) * 4;
    out[o + 1] = tacc;
    out[o + 3] = cacc;
  }

  // Outer-sum rs_i + rs_j via WMMA: D = [rs_i, 1] x [1 ; rs_j]  (16x16x4 f32)
  if (tid < 32) {
    int lane = tid, n = lane & 15, hi = lane >> 4;
    v2f a, bv;
    if (hi == 0) { // lanes 0-15 hold K=0,1 ; lanes 16-31 hold K=2,3 (zeros)
      a.x = rsI[lane]; a.y = 1.0f;
      bv.x = 1.0f;     bv.y = rsJ[lane];
    } else {
      a.x = 0.f; a.y = 0.f; bv.x = 0.f; bv.y = 0.f;
    }
    v8f cz = {0.f, 0.f, 0.f, 0.f, 0.f, 0.f, 0.f, 0.f};
    v8f d = __builtin_amdgcn_wmma_f32_16x16x4_f32(false, a, false, bv,
                                                  (short)0, cz, false, false);
#pragma unroll
    for (int r = 0; r < 8; ++r) {
      int M = r + 8 * hi; // C/D layout: VGPR r = row M (lanes 0-15), M+8 (lanes 16-31)
      float rsum = d[r];  // rs_i[M] + rs_j[n]
      float size_aff = fminf(fmaxf(1.0f - rsum / kImgSize, 0.f), 1.f);
      float xi = boxI[M][0], yi = boxI[M][1], wi = boxI[M][2], hgi = boxI[M][3];
      float xj = boxJ[n][0], yj = boxJ[n][1], wj = boxJ[n][2], hj = boxJ[n][3];
      float mxmin = fminf(xi, xj), mymin = fminf(yi, yj);
      float mxmax = fmaxf(xi + wi - 1.f, xj + wj - 1.f);
      float mymax = fmaxf(yi + hgi - 1.f, yj + hj - 1.f);
      float merged = (mxmax - mxmin + 1.f) * (mymax - mymin + 1.f);
      float fill = fminf(fmaxf(1.0f - (merged - rsum) / kImgSize, 0.f), 1.f);
      size_t o = (((size_t)b * kS + (i0 + M)) * kS + (j0 + n)) * 4;
      out[o + 0] = fill;
      out[o + 2] = size_aff;
    }
  }
}

extern "C" void kernel_launch(void* const* d_in, const int* in_sizes, int n_in,
                              void* d_out, int out_size, void* d_ws, size_t ws_size,
                              hipStream_t stream) {
  (void)in_sizes; (void)n_in; (void)out_size; (void)ws_size;
  const float* imgs  = (const float*)d_in[0];
  const float* grads = (const float*)d_in[1];
  const int*   lab   = (const int*)d_in[2];
  unsigned* ws = (unsigned*)d_ws;
  float* out = (float*)d_out;

  init_ws_kernel<<<(WS_TOTAL + 255) / 256, 256, 0, stream>>>(ws);
  pixel_kernel<<<(kB * kHW) / 256, 256, 0, stream>>>(imgs, lab, ws);
  texture_kernel<<<dim3(kB * kC * kG, 8), 256, 0, stream>>>(grads, lab, ws);
  normalize_kernel<<<(kB * kS) / 8, 256, 0, stream>>>(ws);
  aff_kernel<<<dim3(kS / 16, kS / 16, kB), 256, 0, stream>>>(ws, out);
}